// LocalAttention_8366596293016
// MI455X (gfx1250) — compile-verified
//
#include <hip/hip_runtime.h>
#include <math.h>

// Local bucketed causal attention for MI455X (gfx1250, wave32, WMMA).
// fp32 end-to-end via V_WMMA_F32_16X16X4_F32 (memory-bound problem: ~270MB
// moved vs 8.6 GFLOP -> ~12us floor at 23.3 TB/s; fp32 matrix core is free).

typedef float v2f __attribute__((ext_vector_type(2)));
typedef float v8f __attribute__((ext_vector_type(8)));

#define B_SZ     64
#define T_SZ     4096
#define E_SZ     64
#define BUCKETSZ 64
#define WIN      128
#define NBUCKETS (T_SZ / BUCKETSZ)

// LDS layout (floats). Pitches padded so 16-lane fragment reads hit distinct
// banks ((pitch mod 64) in {1,2,4} => conflict-free) and stay float4/float2
// aligned for staging / ds_load_b64.
#define QPITCH 68
#define KPITCH 68
#define VPITCH 132
#define PPITCH 132
#define QS_OFF 0
#define KS_OFF (BUCKETSZ * QPITCH)              /* 4352  */
#define VT_OFF (KS_OFF + WIN * KPITCH)          /* 13056 */
#define P_OFF  KS_OFF                           /* P overlaps K after GEMM1 */
#define SMEM_FLOATS (VT_OFF + E_SZ * VPITCH)    /* 21504 -> 86016 bytes */

__global__ __launch_bounds__(128, 1)
void local_attn_wmma_f32(const float* __restrict__ q,
                         const float* __restrict__ k,
                         const float* __restrict__ v,
                         float* __restrict__ out) {
  extern __shared__ float smem[];
  float* QS = smem + QS_OFF;
  float* KS = smem + KS_OFF;
  float* VT = smem + VT_OFF;   // V transposed: VT[e][key]
  float* P  = smem + P_OFF;    // exp(S - max), unnormalized

  const int bucket = blockIdx.x;
  const int batch  = blockIdx.y;
  const int tid  = threadIdx.x;
  const int lane = tid & 31;
  const int wave = tid >> 5;
  const int l16  = lane & 15;
  const int half = lane >> 4;      // 0: lanes 0-15, 1: lanes 16-31

  const size_t qbase = ((size_t)batch * T_SZ + (size_t)bucket * BUCKETSZ) * E_SZ;

  // ---- stage Q bucket (64x64) ----
#pragma unroll
  for (int i = 0; i < 8; ++i) {
    int idx = tid + i * 128;                 // 1024 float4s
    int row = idx >> 4;
    int c4  = (idx & 15) << 2;
    float4 val = *(const float4*)(q + qbase + (size_t)row * E_SZ + c4);
    *(float4*)(QS + row * QPITCH + c4) = val;
  }
  // ---- stage K window (128x64) and V window transposed ----
#pragma unroll
  for (int i = 0; i < 16; ++i) {
    int idx = tid + i * 128;                 // 2048 float4s
    int r   = idx >> 4;                      // window row 0..127
    int c4  = (idx & 15) << 2;
    int gr  = bucket * BUCKETSZ - BUCKETSZ + r;
    if (gr < 0) gr = 0;                      // bucket 0: padded keys, masked later
    size_t goff = ((size_t)batch * T_SZ + gr) * E_SZ + c4;
    float4 kv = *(const float4*)(k + goff);
    *(float4*)(KS + r * KPITCH + c4) = kv;
    float4 vv = *(const float4*)(v + goff);
    VT[(c4 + 0) * VPITCH + r] = vv.x;
    VT[(c4 + 1) * VPITCH + r] = vv.y;
    VT[(c4 + 2) * VPITCH + r] = vv.z;
    VT[(c4 + 3) * VPITCH + r] = vv.w;
  }
  __syncthreads();

  const int qr0 = wave * 16;                 // this wave's query-row tile

  // ---- hoist Q A-fragments: 16x64 tile -> 16 k-steps of 16x4 ----
  // A layout (32-bit 16x4): lane L holds M=L%16, K=2*(L/16)+{0,1}
  v2f qa[16];
#pragma unroll
  for (int kk = 0; kk < 16; ++kk)
    qa[kk] = *(const v2f*)(QS + (qr0 + l16) * QPITCH + kk * 4 + half * 2);

  // ---- GEMM1: S strip (16x128) = Q(16x64) x K^T, 8 column tiles ----
  // B layout (32-bit 4x16): lane L holds N=L%16, K=2*(L/16)+{0,1}
  v8f sa[8];
#pragma unroll
  for (int j = 0; j < 8; ++j) {
    v8f c = {0.f, 0.f, 0.f, 0.f, 0.f, 0.f, 0.f, 0.f};
#pragma unroll
    for (int kk = 0; kk < 16; ++kk) {
      v2f b = *(const v2f*)(KS + (j * 16 + l16) * KPITCH + kk * 4 + half * 2);
      c = __builtin_amdgcn_wmma_f32_16x16x4_f32(false, qa[kk], false, b,
                                                (short)0, c, false, false);
    }
    sa[j] = c;
  }
  __syncthreads();   // all waves finished reading KS; P may now overwrite it

  // ---- scale + masks + running row max (C layout: VGPR r -> row r / r+8) ----
  float rm[8];
#pragma unroll
  for (int r = 0; r < 8; ++r) rm[r] = -3.0e38f;
#pragma unroll
  for (int j = 0; j < 8; ++j) {
#pragma unroll
    for (int r = 0; r < 8; ++r) {
      int row = qr0 + r + half * 8;          // query pos within bucket
      int key = j * 16 + l16;                // window pos 0..127
      float s = sa[j][r] * 0.125f;           // E^-0.5
      bool masked = (key < BUCKETSZ) ? (bucket == 0)          // pad mask
                                     : (row < key - BUCKETSZ); // causal
      s = masked ? -1.0e30f : s;
      sa[j][r] = s;
      rm[r] = fmaxf(rm[r], s);
    }
  }
#pragma unroll
  for (int r = 0; r < 8; ++r) {
    float m = rm[r];
#pragma unroll
    for (int msk = 1; msk < 16; msk <<= 1) m = fmaxf(m, __shfl_xor(m, msk, 32));
    rm[r] = m;
  }

  // ---- exp, row sum, store unnormalized P to LDS ----
  float rs[8];
#pragma unroll
  for (int r = 0; r < 8; ++r) rs[r] = 0.0f;
#pragma unroll
  for (int j = 0; j < 8; ++j) {
#pragma unroll
    for (int r = 0; r < 8; ++r) {
      int row = qr0 + r + half * 8;
      float e = __expf(sa[j][r] - rm[r]);
      rs[r] += e;
      P[row * PPITCH + j * 16 + l16] = e;
    }
  }
  float inv[8];
#pragma unroll
  for (int r = 0; r < 8; ++r) {
    float ssum = rs[r];
#pragma unroll
    for (int msk = 1; msk < 16; msk <<= 1) ssum += __shfl_xor(ssum, msk, 32);
    inv[r] = 1.0f / ssum;
  }
  __syncthreads();   // P stores visible (also keeps waves in step)

  // ---- GEMM2: O (16x64) = P(16x128) x V(128x64), 4 e-tiles, 32 k-steps ----
  v8f oacc[4];
#pragma unroll
  for (int n = 0; n < 4; ++n)
    oacc[n] = (v8f){0.f, 0.f, 0.f, 0.f, 0.f, 0.f, 0.f, 0.f};
#pragma unroll
  for (int kk = 0; kk < 32; ++kk) {
    v2f a = *(const v2f*)(P + (qr0 + l16) * PPITCH + kk * 4 + half * 2);
#pragma unroll
    for (int n = 0; n < 4; ++n) {
      v2f b = *(const v2f*)(VT + (n * 16 + l16) * VPITCH + kk * 4 + half * 2);
      oacc[n] = __builtin_amdgcn_wmma_f32_16x16x4_f32(false, a, false, b,
                                                      (short)0, oacc[n],
                                                      false, false);
    }
  }

  // ---- fold softmax normalization into output, store ----
#pragma unroll
  for (int n = 0; n < 4; ++n) {
#pragma unroll
    for (int r = 0; r < 8; ++r) {
      int row = qr0 + r + half * 8;
      out[qbase + (size_t)row * E_SZ + n * 16 + l16] = oacc[n][r] * inv[r];
    }
  }
}

extern "C" void kernel_launch(void* const* d_in, const int* in_sizes, int n_in,
                              void* d_out, int out_size, void* d_ws, size_t ws_size,
                              hipStream_t stream) {
  (void)in_sizes; (void)n_in; (void)d_ws; (void)ws_size; (void)out_size;
  const float* q = (const float*)d_in[0];
  const float* k = (const float*)d_in[1];
  const float* v = (const float*)d_in[2];
  float* out = (float*)d_out;

  dim3 grid(NBUCKETS, B_SZ);   // (bucket, batch)
  dim3 block(128);             // 4 waves of 32
  size_t smem = (size_t)SMEM_FLOATS * sizeof(float);  // ~84 KB (gfx1250: 320KB/WG)
  local_attn_wmma_f32<<<grid, block, smem, stream>>>(q, k, v, out);
}